// CriterionOhemDSN_72430328480962
// MI455X (gfx1250) — compile-verified
//
#include <hip/hip_runtime.h>
#include <hip/hip_bf16.h>

// ---------------------------------------------------------------------------
// CriterionOhemDSN for MI455X (gfx1250, wave32).
// Fully fused: bilinear 97->769 upsample + log-softmax + p_true in one pass
// (never materializes the 172MB upsampled logits), exact OHEM k-th-smallest
// via 4-level radix select on float bits, weighted CE reduction finished with
// V_WMMA_F32_16X16X4_F32 (f32 matrix path). LDS staging uses the gfx1250
// async global->LDS path (ASYNCcnt) when the builtins are available.
// ---------------------------------------------------------------------------

#define H_IN   97
#define W_IN   97
#define NCLS   19
#define H_OUT  769
#define W_OUT  769
#define NBATCH 4
#define NPIX   (NBATCH * H_OUT * W_OUT)   // 2,365,444
#define MIN_KEPT 100000
#define THRESH 0.7f
#define FINAL_BLOCKS 512

#if defined(__AMDGCN__) &&                                                  \
    __has_builtin(__builtin_amdgcn_global_load_async_to_lds_b32) &&         \
    __has_builtin(__builtin_amdgcn_s_wait_asynccnt)
#define OHEM_ASYNC_LDS 1
#else
#define OHEM_ASYNC_LDS 0
#endif

#if OHEM_ASYNC_LDS
typedef __attribute__((address_space(1))) int gas_i32;  // global b32
typedef __attribute__((address_space(3))) int las_i32;  // LDS b32
#endif

__constant__ float c_w[NCLS] = {
    0.8373f, 0.918f, 0.866f, 1.0345f, 1.0166f, 0.9969f, 0.9754f,
    1.0489f, 0.8786f, 1.0023f, 0.9539f, 0.9843f, 1.1116f, 0.9037f,
    1.0865f, 1.0955f, 1.0865f, 1.1529f, 1.0507f};

typedef float v2f __attribute__((ext_vector_type(2)));
typedef float v8f __attribute__((ext_vector_type(8)));

// ---------------------------------------------------------------------------
// init: zero radix histogram + control block (prefix, rank)
// ---------------------------------------------------------------------------
__global__ void ohem_init_kernel(unsigned* __restrict__ hist,
                                 unsigned* __restrict__ ctl) {
  int t = threadIdx.x;
  if (t < 256) hist[t] = 0u;
  if (t == 0) {
    ctl[0] = 0u;                 // known-prefix of k-th smallest (high bits)
    ctl[1] = (unsigned)MIN_KEPT; // remaining rank (1-based)
    ctl[2] = 0u;                 // threshold bits (filled by last scan)
  }
}

// ---------------------------------------------------------------------------
// pass1: one block per (output row y, batch n).
// Stage the two source rows (all 19 channels) into LDS (async global->LDS on
// gfx1250), then for each of the 769 output pixels: bilinear interp of 19
// logits, log-softmax, p_true. Writes p bits + log p per pixel and builds the
// top-byte radix histogram (fused radix level 0).
// ---------------------------------------------------------------------------
__global__ void ohem_pass1_kernel(const float* __restrict__ logits,
                                  const int* __restrict__ target,
                                  unsigned* __restrict__ pbits,
                                  float* __restrict__ lp_arr,
                                  unsigned* __restrict__ hist) {
  __shared__ float sh[2][NCLS][W_IN + 3];  // pad to dodge bank conflicts
  __shared__ unsigned shist[256];

  const int y = blockIdx.x;
  const int n = blockIdx.y;
  const int tid = threadIdx.x;

  for (int i = tid; i < 256; i += blockDim.x) shist[i] = 0u;

  // torch align_corners=False / jax half-pixel bilinear source coordinate
  float sy = (y + 0.5f) * ((float)H_IN / (float)H_OUT) - 0.5f;
  sy = fmaxf(sy, 0.0f);
  float fyf = floorf(sy);
  const float fy = sy - fyf;
  int y0 = (int)fyf;
  int y1 = y0 + 1;
  y0 = min(max(y0, 0), H_IN - 1);
  y1 = min(max(y1, 0), H_IN - 1);

  for (int i = tid; i < 2 * NCLS * W_IN; i += blockDim.x) {
    int x  = i % W_IN;
    int rc = i / W_IN;
    int r  = rc & 1;
    int c  = rc >> 1;
    int yy = r ? y1 : y0;
    const float* gsrc = &logits[(((size_t)n * NCLS + c) * H_IN + yy) * W_IN + x];
#if OHEM_ASYNC_LDS
    // gfx1250 async DMA: global -> LDS, tracked by ASYNCcnt (no VGPR dest)
    __builtin_amdgcn_global_load_async_to_lds_b32(
        (gas_i32*)gsrc, (las_i32*)&sh[r][c][x], 0, 0);
#else
    sh[r][c][x] = *gsrc;
#endif
  }
#if OHEM_ASYNC_LDS
  __builtin_amdgcn_s_wait_asynccnt(0);
#endif
  __syncthreads();

  const float wy1 = fy, wy0 = 1.0f - fy;

  for (int x = tid; x < W_OUT; x += blockDim.x) {
    float sx = (x + 0.5f) * ((float)W_IN / (float)W_OUT) - 0.5f;
    sx = fmaxf(sx, 0.0f);
    float fxf = floorf(sx);
    const float fx = sx - fxf;
    int x0 = (int)fxf;
    int x1 = x0 + 1;
    x0 = min(max(x0, 0), W_IN - 1);
    x1 = min(max(x1, 0), W_IN - 1);
    const float wx1 = fx, wx0 = 1.0f - fx;

    const int idx = (n * H_OUT + y) * W_OUT + x;
    const int t = target[idx];

    float vals[NCLS];
    float m = -3.402823466e38f;
    float vt = 0.0f;
#pragma unroll
    for (int c = 0; c < NCLS; ++c) {
      float v0 = sh[0][c][x0] * wx0 + sh[0][c][x1] * wx1;
      float v1 = sh[1][c][x0] * wx0 + sh[1][c][x1] * wx1;
      float v  = wy0 * v0 + wy1 * v1;
      vals[c] = v;
      m = fmaxf(m, v);
      vt = (c == t) ? v : vt;
    }
    float s = 0.0f;
#pragma unroll
    for (int c = 0; c < NCLS; ++c) s += expf(vals[c] - m);
    const float lse = m + logf(s);

    float lp;
    unsigned pb;
    if (t >= 0 && t < NCLS) {
      lp = vt - lse;               // log p_true
      pb = __float_as_uint(expf(lp));
    } else {
      lp = 0.0f;
      pb = 0x7f800000u;            // ignore label -> +inf, never kept
    }
    pbits[idx] = pb;
    lp_arr[idx] = lp;
    atomicAdd(&shist[pb >> 24], 1u);
  }
  __syncthreads();
  for (int i = tid; i < 256; i += blockDim.x)
    if (shist[i]) atomicAdd(&hist[i], shist[i]);
}

// ---------------------------------------------------------------------------
// radix scan: pick the byte bucket containing the remaining rank, update
// prefix/rank, zero the histogram for the next level. On shift==0 the prefix
// is the exact bit pattern of the k-th smallest p; threshold = max(kth, 0.7).
// ---------------------------------------------------------------------------
__global__ void ohem_radix_scan_kernel(unsigned* __restrict__ hist,
                                       unsigned* __restrict__ ctl, int shift) {
  __shared__ unsigned h[256];
  int t = threadIdx.x;
  h[t] = hist[t];
  __syncthreads();
  if (t == 0) {
    unsigned rank = ctl[1];
    unsigned prefix = ctl[0];
    unsigned cum = 0;
    for (int b = 0; b < 256; ++b) {
      unsigned c = h[b];
      if (cum + c >= rank) {
        prefix |= ((unsigned)b) << shift;
        rank -= cum;
        break;
      }
      cum += c;
    }
    ctl[0] = prefix;
    ctl[1] = rank;
    if (shift == 0) {
      float kth = __uint_as_float(prefix);
      ctl[2] = __float_as_uint(fmaxf(kth, THRESH));
    }
  }
  __syncthreads();
  hist[t] = 0u;
}

// ---------------------------------------------------------------------------
// radix histogram over next byte, restricted to keys matching current prefix
// ---------------------------------------------------------------------------
__global__ void ohem_radix_hist_kernel(const unsigned* __restrict__ pbits,
                                       const unsigned* __restrict__ ctl,
                                       unsigned* __restrict__ hist, int shift) {
  __shared__ unsigned shist[256];
  int t = threadIdx.x;
  shist[t] = 0u;
  __syncthreads();
  const unsigned prefix = ctl[0];
  const int hs = shift + 8;
  for (int i = blockIdx.x * blockDim.x + t; i < NPIX;
       i += gridDim.x * blockDim.x) {
    unsigned k = pbits[i];
    if (((k ^ prefix) >> hs) == 0u)
      atomicAdd(&shist[(k >> shift) & 255u], 1u);
  }
  __syncthreads();
  if (shist[t]) atomicAdd(&hist[t], shist[t]);
}

// ---------------------------------------------------------------------------
// final pass: weighted CE sums over kept pixels (p <= threshold)
// ---------------------------------------------------------------------------
__global__ void ohem_final_kernel(const unsigned* __restrict__ pbits,
                                  const float* __restrict__ lp_arr,
                                  const int* __restrict__ target,
                                  const unsigned* __restrict__ ctl,
                                  float* __restrict__ partials) {
  __shared__ float s1[256];
  __shared__ float s2[256];
  const float thr = __uint_as_float(ctl[2]);
  int t = threadIdx.x;
  float a = 0.0f, b = 0.0f;
  for (int i = blockIdx.x * blockDim.x + t; i < NPIX;
       i += gridDim.x * blockDim.x) {
    float p = __uint_as_float(pbits[i]);
    if (p <= thr) {
      int tg = target[i];
      if (tg >= 0 && tg < NCLS) {
        float w = c_w[tg];
        a += w * (-lp_arr[i]);
        b += w;
      }
    }
  }
  s1[t] = a;
  s2[t] = b;
  __syncthreads();
  for (int s = 128; s > 0; s >>= 1) {
    if (t < s) {
      s1[t] += s1[t + s];
      s2[t] += s2[t + s];
    }
    __syncthreads();
  }
  if (t == 0) {
    partials[blockIdx.x] = s1[0];
    partials[FINAL_BLOCKS + blockIdx.x] = s2[0];
  }
}

// ---------------------------------------------------------------------------
// WMMA reduction: one wave (32 lanes, EXEC all ones as required). Each
// iteration packs 64 partials as a 16x4 f32 A-matrix, multiplies by an
// all-ones 4x16 B, accumulating in C. Since B==1, sum(C) == 16 * sum(A);
// the 16x factor cancels in the ratio sum(wce)/sum(w).
// A 16x4 f32 layout (ISA 7.12.2): lanes 0-15 hold (M=lane, K=0/1) in
// VGPR0/1; lanes 16-31 hold (M=lane-16, K=2/3).
// ---------------------------------------------------------------------------
__global__ void ohem_wmma_reduce_kernel(const float* __restrict__ partials,
                                        float* __restrict__ loss_out) {
  const int lane = threadIdx.x;  // 0..31, one full wave32
  const int base = (lane < 16) ? lane * 4 : (lane - 16) * 4 + 2;

  v8f acc_a = {};
  v8f acc_b = {};
  v2f ones;
  ones.x = 1.0f;
  ones.y = 1.0f;

  for (int chunk = 0; chunk < FINAL_BLOCKS; chunk += 64) {
    v2f av, bv;
    av.x = partials[chunk + base];
    av.y = partials[chunk + base + 1];
    bv.x = partials[FINAL_BLOCKS + chunk + base];
    bv.y = partials[FINAL_BLOCKS + chunk + base + 1];
    acc_a = __builtin_amdgcn_wmma_f32_16x16x4_f32(
        false, av, false, ones, (short)0, acc_a, false, false);
    acc_b = __builtin_amdgcn_wmma_f32_16x16x4_f32(
        false, bv, false, ones, (short)0, acc_b, false, false);
  }

  float sa = 0.0f, sb = 0.0f;
#pragma unroll
  for (int r = 0; r < 8; ++r) {
    sa += acc_a[r];
    sb += acc_b[r];
  }
  // full cross-wave butterfly: sum over all (M,N) entries of C
  for (int off = 16; off > 0; off >>= 1) {
    sa += __shfl_xor(sa, off, 32);
    sb += __shfl_xor(sb, off, 32);
  }
  if (lane == 0) loss_out[0] = sa / sb;  // 16x redundancy cancels
}

__global__ void ohem_combine_kernel(const float* __restrict__ loss_buf,
                                    float* __restrict__ out) {
  out[0] = 0.4f * loss_buf[0] + loss_buf[1];
}

// ---------------------------------------------------------------------------
static inline size_t align_up(size_t v, size_t a) { return (v + a - 1) & ~(a - 1); }

extern "C" void kernel_launch(void* const* d_in, const int* in_sizes, int n_in,
                              void* d_out, int out_size, void* d_ws,
                              size_t ws_size, hipStream_t stream) {
  const float* pred1 = (const float*)d_in[0];
  const float* pred2 = (const float*)d_in[1];
  const int* target = (const int*)d_in[2];
  float* out = (float*)d_out;

  char* ws = (char*)d_ws;
  size_t off = 0;
  unsigned* pbits = (unsigned*)(ws + off);
  off = align_up(off + (size_t)NPIX * sizeof(unsigned), 256);
  float* lp_arr = (float*)(ws + off);
  off = align_up(off + (size_t)NPIX * sizeof(float), 256);
  unsigned* hist = (unsigned*)(ws + off);
  off = align_up(off + 256 * sizeof(unsigned), 256);
  unsigned* ctl = (unsigned*)(ws + off);
  off = align_up(off + 16, 256);
  float* partials = (float*)(ws + off);
  off = align_up(off + 2 * FINAL_BLOCKS * sizeof(float), 256);
  float* loss_buf = (float*)(ws + off);

  const float* preds[2] = {pred1, pred2};
  for (int p = 0; p < 2; ++p) {
    ohem_init_kernel<<<1, 256, 0, stream>>>(hist, ctl);
    ohem_pass1_kernel<<<dim3(H_OUT, NBATCH), 256, 0, stream>>>(
        preds[p], target, pbits, lp_arr, hist);
    ohem_radix_scan_kernel<<<1, 256, 0, stream>>>(hist, ctl, 24);
    ohem_radix_hist_kernel<<<1024, 256, 0, stream>>>(pbits, ctl, hist, 16);
    ohem_radix_scan_kernel<<<1, 256, 0, stream>>>(hist, ctl, 16);
    ohem_radix_hist_kernel<<<1024, 256, 0, stream>>>(pbits, ctl, hist, 8);
    ohem_radix_scan_kernel<<<1, 256, 0, stream>>>(hist, ctl, 8);
    ohem_radix_hist_kernel<<<1024, 256, 0, stream>>>(pbits, ctl, hist, 0);
    ohem_radix_scan_kernel<<<1, 256, 0, stream>>>(hist, ctl, 0);
    ohem_final_kernel<<<FINAL_BLOCKS, 256, 0, stream>>>(pbits, lp_arr, target,
                                                        ctl, partials);
    ohem_wmma_reduce_kernel<<<1, 32, 0, stream>>>(partials, loss_buf + p);
  }
  ohem_combine_kernel<<<1, 1, 0, stream>>>(loss_buf, out);
}